// IndexTTS_E_65206193488317
// MI455X (gfx1250) — compile-verified
//
#include <hip/hip_runtime.h>

#define LYR  24
#define NH   16
#define HD   64
#define DM   1024
#define FF   4096
#define VOC  8194
#define PAST 2047
#define KVL  2048

typedef __attribute__((ext_vector_type(2))) float v2f;
typedef __attribute__((ext_vector_type(4))) float v4f;
typedef __attribute__((ext_vector_type(8))) float v8f;

// ---------------- output layout (floats) ----------------
#define OUT_KS      ((size_t)0)
#define OUT_VS      ((size_t)LYR * NH * HD * KVL)              // 50331648
#define OUT_KVLEN   (OUT_VS + (size_t)LYR * NH * KVL * HD)     // 100663296
#define OUT_LAST    (OUT_KVLEN + 1)
#define OUT_MAXIDS  (OUT_LAST + DM)

// ---------------- workspace layout (floats) -------------
#define WS_H      0
#define WS_X      1024
#define WS_Q      2048
#define WS_ATTN   3072
#define WS_Y      4096
#define WS_LOGITS 8192

// =========================================================
// Core WMMA matvec helper: accumulate y[col] = sum_k x[k]*W[k][col]
// Wcol = &W[0][col] (per-lane column already applied, lanes 16-31 use
// the same column as lanes 0-15). stride = row stride in floats.
// Result (M=0 row of D) is in c[0] of lanes 0..15.
// A layout (16x4 f32): VGPR0 = {K0 lanes0-15, K2 lanes16-31},
//                      VGPR1 = {K1, K3}; only row M=0 nonzero.
// B layout (4x16 f32): VGPR0 = {K0-row lanes0-15, K2-row lanes16-31},
//                      VGPR1 = {K1-row, K3-row}.
// =========================================================
__device__ __forceinline__ float wmma_colaccum(const float* __restrict__ Wcol,
                                               size_t stride,
                                               const float* __restrict__ xl,
                                               int K)
{
    const int lane  = threadIdx.x & 31;
    const int half  = lane >> 4;          // 0: K0/K1 rows, 1: K2/K3 rows
    const bool a_en = ((lane & 15) == 0); // lanes 0 and 16 carry the x row
    v8f c = {};
    const float* wp = Wcol + (size_t)(2 * half) * stride;
    #pragma unroll 4
    for (int k = 0; k < K; k += 4) {
        v2f a, b;
        float x0 = xl[k + 2 * half];
        float x1 = xl[k + 2 * half + 1];
        a.x = a_en ? x0 : 0.0f;
        a.y = a_en ? x1 : 0.0f;
        b.x = wp[0];
        b.y = wp[stride];
        c = __builtin_amdgcn_wmma_f32_16x16x4_f32(false, a, false, b,
                                                  (short)0, c, false, false);
        wp += 4 * stride;
    }
    return c[0];
}

// =========================================================
// KV cache concat copies
// =========================================================
__global__ void k_copy_pastk(const float* __restrict__ pk, float* __restrict__ ok)
{
    // one block per (l,h,e) row: copy 2047 floats into a 2048-stride row
    size_t row = blockIdx.x;
    const float* s = pk + row * (size_t)PAST;
    float*       d = ok + row * (size_t)KVL;
    for (int t = threadIdx.x; t < PAST; t += blockDim.x) d[t] = s[t];
}

__global__ void k_copy_pastv(const float* __restrict__ pv, float* __restrict__ ov)
{
    // flat float4 copy with per-(l,h) row remap: out = in + lh*HD floats
    const size_t per = (size_t)PAST * HD / 4;   // v4f per (l,h)
    const size_t n   = (size_t)LYR * NH * per;
    const v4f* s = (const v4f*)pv;
    v4f*       d = (v4f*)ov;
    size_t idx    = (size_t)blockIdx.x * blockDim.x + threadIdx.x;
    size_t stride = (size_t)gridDim.x * blockDim.x;
    for (; idx < n; idx += stride) {
        size_t lh = idx / per;
        d[idx + lh * (HD / 4)] = s[idx];
    }
}

__global__ void k_copy_h(const float* __restrict__ src, float* __restrict__ dst)
{
    for (int i = threadIdx.x; i < DM; i += blockDim.x) dst[i] = src[i];
}

// =========================================================
// LayerNorm (1 block, 256 threads, D=1024)
// =========================================================
__global__ void k_layernorm(const float* __restrict__ src,
                            const float* __restrict__ g,
                            const float* __restrict__ b,
                            float* __restrict__ dst,
                            float* __restrict__ dst2)
{
    __shared__ float s_sum[256], s_sq[256], s_mr[2];
    int tid = threadIdx.x;
    float v[4], sum = 0.f, sq = 0.f;
    #pragma unroll
    for (int i = 0; i < 4; i++) {
        v[i] = src[tid + i * 256];
        sum += v[i]; sq += v[i] * v[i];
    }
    s_sum[tid] = sum; s_sq[tid] = sq;
    __syncthreads();
    for (int off = 128; off > 0; off >>= 1) {
        if (tid < off) { s_sum[tid] += s_sum[tid + off]; s_sq[tid] += s_sq[tid + off]; }
        __syncthreads();
    }
    if (tid == 0) {
        float m   = s_sum[0] * (1.0f / DM);
        float var = s_sq[0] * (1.0f / DM) - m * m;
        s_mr[0] = m; s_mr[1] = rsqrtf(var + 1e-5f);
    }
    __syncthreads();
    float m = s_mr[0], r = s_mr[1];
    #pragma unroll
    for (int i = 0; i < 4; i++) {
        int d = tid + i * 256;
        float o = (v[i] - m) * r * g[d] + b[d];
        dst[d] = o;
        if (dst2) dst2[d] = o;
    }
}

// =========================================================
// QKV projection: 192 waves (3 mats * 16 heads * 4 n-tiles)
// =========================================================
__global__ void k_qkv(const float* __restrict__ x,
                      const float* __restrict__ Wq, const float* __restrict__ bq,
                      const float* __restrict__ Wk, const float* __restrict__ bk,
                      const float* __restrict__ Wv, const float* __restrict__ bv,
                      float* __restrict__ q_ws,
                      float* __restrict__ out_ks, float* __restrict__ out_vs,
                      int layer)
{
    __shared__ float xl[DM];
    for (int i = threadIdx.x; i < DM; i += blockDim.x) xl[i] = x[i];
    __syncthreads();

    int wid  = (blockIdx.x * blockDim.x + threadIdx.x) >> 5;   // 0..191
    int lane = threadIdx.x & 31;
    int m    = wid >> 6;        // 0=q 1=k 2=v
    int rem  = wid & 63;
    int h    = rem >> 2;
    int n0   = (rem & 3) * 16;

    const float* W    = (m == 0 ? Wq : m == 1 ? Wk : Wv) + (size_t)h * DM * HD;
    const float* bias = (m == 0 ? bq : m == 1 ? bk : bv) + (size_t)h * HD;
    int col = n0 + (lane & 15);

    float r = wmma_colaccum(W + col, HD, xl, DM);

    if (lane < 16) {
        float val = r + bias[col];
        size_t lh = (size_t)layer * NH + h;
        if (m == 0)      q_ws[h * HD + col] = val;
        else if (m == 1) out_ks[(lh * HD + col) * KVL + (KVL - 1)] = val;
        else             out_vs[(lh * KVL + (KVL - 1)) * HD + col] = val;
    }
}

// =========================================================
// Attention: one block (512 thr) per head, kv = 2048
// =========================================================
__global__ void k_attn(const float* __restrict__ q_ws,
                       const float* __restrict__ mask_scale,
                       const float* __restrict__ ks, const float* __restrict__ vs,
                       float* __restrict__ attn_ws, int layer)
{
    __shared__ float sq[HD];
    __shared__ float sc[KVL];
    __shared__ float red[512];
    int h = blockIdx.x, tid = threadIdx.x;
    size_t lh = (size_t)layer * NH + h;

    if (tid < HD) sq[tid] = q_ws[h * HD + tid];
    __syncthreads();

    float msc = -128.0f * mask_scale[0];
    const float* kb = ks + lh * (size_t)HD * KVL;
    for (int t = tid; t < KVL; t += 512) {
        float s = 0.f;
        const float* kp = kb + t;
        #pragma unroll 8
        for (int e = 0; e < HD; e++) s += sq[e] * kp[(size_t)e * KVL];
        if (t > 0) s += msc;
        sc[t] = s;
    }
    __syncthreads();

    // max
    float mx = -3.402823466e38f;
    for (int t = tid; t < KVL; t += 512) mx = fmaxf(mx, sc[t]);
    red[tid] = mx; __syncthreads();
    for (int off = 256; off > 0; off >>= 1) {
        if (tid < off) red[tid] = fmaxf(red[tid], red[tid + off]);
        __syncthreads();
    }
    mx = red[0]; __syncthreads();

    // exp + sum
    float sm = 0.f;
    for (int t = tid; t < KVL; t += 512) { float e = expf(sc[t] - mx); sc[t] = e; sm += e; }
    red[tid] = sm; __syncthreads();
    for (int off = 256; off > 0; off >>= 1) {
        if (tid < off) red[tid] += red[tid + off];
        __syncthreads();
    }
    float inv = 1.0f / red[0];
    __syncthreads();

    // probs @ V : thread = (e = tid&63, tg = tid>>6)
    int e = tid & 63, tg = tid >> 6;
    const float* vb = vs + lh * (size_t)KVL * HD;
    float acc = 0.f;
    for (int t = tg; t < KVL; t += 8) acc += sc[t] * vb[(size_t)t * HD + e];
    red[tid] = acc; __syncthreads();
    if (tid < HD) {
        float a = 0.f;
        #pragma unroll
        for (int g2 = 0; g2 < 8; g2++) a += red[g2 * 64 + tid];
        attn_ws[h * HD + tid] = a * inv;
    }
}

// =========================================================
// Out-projection + residual: 64 waves over N=1024
// =========================================================
__global__ void k_oproj(const float* __restrict__ attn_ws, const float* __restrict__ Wo,
                        const float* __restrict__ bo, float* __restrict__ hbuf)
{
    __shared__ float xl[DM];
    for (int i = threadIdx.x; i < DM; i += blockDim.x) xl[i] = attn_ws[i];
    __syncthreads();
    int wid = (blockIdx.x * blockDim.x + threadIdx.x) >> 5;    // 0..63
    int lane = threadIdx.x & 31;
    int col = wid * 16 + (lane & 15);
    float r = wmma_colaccum(Wo + col, DM, xl, DM);
    if (lane < 16) hbuf[col] = hbuf[col] + r + bo[col];
}

// =========================================================
// FC + tanh-GELU: 256 waves over N=4096
// =========================================================
__global__ void k_fc(const float* __restrict__ x, const float* __restrict__ Wfc,
                     const float* __restrict__ bfc, float* __restrict__ y)
{
    __shared__ float xl[DM];
    for (int i = threadIdx.x; i < DM; i += blockDim.x) xl[i] = x[i];
    __syncthreads();
    int wid = (blockIdx.x * blockDim.x + threadIdx.x) >> 5;    // 0..255
    int lane = threadIdx.x & 31;
    int col = wid * 16 + (lane & 15);
    float r = wmma_colaccum(Wfc + col, FF, xl, DM);
    if (lane < 16) {
        r += bfc[col];
        float t = r + 0.044715f * r * r * r;
        y[col] = 0.5f * r * (1.0f + tanhf(0.7978845608028654f * t));
    }
}

// =========================================================
// MLP projection + residual: 64 waves, K=4096
// =========================================================
__global__ void k_mp(const float* __restrict__ y, const float* __restrict__ Wmp,
                     const float* __restrict__ bmp, float* __restrict__ hbuf)
{
    __shared__ float xl[FF];
    for (int i = threadIdx.x; i < FF; i += blockDim.x) xl[i] = y[i];
    __syncthreads();
    int wid = (blockIdx.x * blockDim.x + threadIdx.x) >> 5;    // 0..63
    int lane = threadIdx.x & 31;
    int col = wid * 16 + (lane & 15);
    float r = wmma_colaccum(Wmp + col, DM, xl, FF);
    if (lane < 16) hbuf[col] = hbuf[col] + r + bmp[col];
}

// =========================================================
// Logits: 513 tiles over V=8194 (tail clamped)
// =========================================================
__global__ void k_logits(const float* __restrict__ x, const float* __restrict__ lm_w,
                         const float* __restrict__ scale, float* __restrict__ logits)
{
    __shared__ float xl[DM];
    for (int i = threadIdx.x; i < DM; i += blockDim.x) xl[i] = x[i];
    __syncthreads();
    int wid = (blockIdx.x * blockDim.x + threadIdx.x) >> 5;
    int lane = threadIdx.x & 31;
    if (wid < (VOC + 15) / 16) {            // wave-uniform guard (EXEC stays all-1)
        int col  = wid * 16 + (lane & 15);
        int colc = col < VOC ? col : VOC - 1;
        float r = wmma_colaccum(lm_w + colc, VOC, xl, DM);
        if (lane < 16 && col < VOC) logits[col] = r * scale[col];
    }
}

// =========================================================
// Argmax + int32 outputs
// =========================================================
__global__ void k_finalize(const float* __restrict__ logits, float* __restrict__ out)
{
    __shared__ float bv[256];
    __shared__ int   bi[256];
    int tid = threadIdx.x;
    float best = -3.402823466e38f; int bidx = 0;
    for (int v = tid; v < VOC; v += 256) {
        float f = logits[v];
        if (f > best) { best = f; bidx = v; }
    }
    bv[tid] = best; bi[tid] = bidx;
    __syncthreads();
    for (int off = 128; off > 0; off >>= 1) {
        if (tid < off) {
            if (bv[tid + off] > bv[tid] ||
                (bv[tid + off] == bv[tid] && bi[tid + off] < bi[tid])) {
                bv[tid] = bv[tid + off]; bi[tid] = bi[tid + off];
            }
        }
        __syncthreads();
    }
    if (tid == 0) {
        ((int*)out)[OUT_KVLEN]  = KVL;      // int32 bits in float slot
        ((int*)out)[OUT_MAXIDS] = bi[0];
    }
}

// =========================================================
extern "C" void kernel_launch(void* const* d_in, const int* in_sizes, int n_in,
                              void* d_out, int out_size, void* d_ws, size_t ws_size,
                              hipStream_t stream)
{
    (void)in_sizes; (void)n_in; (void)out_size; (void)ws_size;
    const float* past_k  = (const float*)d_in[0];
    const float* past_v  = (const float*)d_in[1];
    const float* hidden  = (const float*)d_in[2];
    const float* lscale  = (const float*)d_in[3];
    const float* mscale  = (const float*)d_in[4];
    const float* ln1_g   = (const float*)d_in[5];
    const float* ln1_b   = (const float*)d_in[6];
    const float* Wq      = (const float*)d_in[7];
    const float* bq      = (const float*)d_in[8];
    const float* Wk      = (const float*)d_in[9];
    const float* bk      = (const float*)d_in[10];
    const float* Wv      = (const float*)d_in[11];
    const float* bv      = (const float*)d_in[12];
    const float* Wo      = (const float*)d_in[13];
    const float* bo      = (const float*)d_in[14];
    const float* ln2_g   = (const float*)d_in[15];
    const float* ln2_b   = (const float*)d_in[16];
    const float* Wfc     = (const float*)d_in[17];
    const float* bfc     = (const float*)d_in[18];
    const float* Wmp     = (const float*)d_in[19];
    const float* bmp     = (const float*)d_in[20];
    const float* lnf_g   = (const float*)d_in[21];
    const float* lnf_b   = (const float*)d_in[22];
    const float* lm_w    = (const float*)d_in[23];

    float* out    = (float*)d_out;
    float* out_ks = out + OUT_KS;
    float* out_vs = out + OUT_VS;
    float* ws     = (float*)d_ws;

    // KV cache concat (independent of compute chain)
    k_copy_pastk<<<LYR * NH * HD, 256, 0, stream>>>(past_k, out_ks);
    k_copy_pastv<<<8192, 256, 0, stream>>>(past_v, out_vs);
    k_copy_h<<<1, 256, 0, stream>>>(hidden, ws + WS_H);

    for (int l = 0; l < LYR; l++) {
        k_layernorm<<<1, 256, 0, stream>>>(ws + WS_H, ln1_g + (size_t)l * DM,
                                           ln1_b + (size_t)l * DM, ws + WS_X, nullptr);
        k_qkv<<<48, 128, 0, stream>>>(ws + WS_X,
            Wq + (size_t)l * NH * DM * HD, bq + (size_t)l * NH * HD,
            Wk + (size_t)l * NH * DM * HD, bk + (size_t)l * NH * HD,
            Wv + (size_t)l * NH * DM * HD, bv + (size_t)l * NH * HD,
            ws + WS_Q, out_ks, out_vs, l);
        k_attn<<<NH, 512, 0, stream>>>(ws + WS_Q, mscale, out_ks, out_vs,
                                       ws + WS_ATTN, l);
        k_oproj<<<16, 128, 0, stream>>>(ws + WS_ATTN,
            Wo + (size_t)l * NH * HD * DM, bo + (size_t)l * DM, ws + WS_H);
        k_layernorm<<<1, 256, 0, stream>>>(ws + WS_H, ln2_g + (size_t)l * DM,
                                           ln2_b + (size_t)l * DM, ws + WS_X, nullptr);
        k_fc<<<64, 128, 0, stream>>>(ws + WS_X, Wfc + (size_t)l * DM * FF,
                                     bfc + (size_t)l * FF, ws + WS_Y);
        k_mp<<<16, 128, 0, stream>>>(ws + WS_Y, Wmp + (size_t)l * FF * DM,
                                     bmp + (size_t)l * DM, ws + WS_H);
    }

    // final LN writes `last` to d_out and x to ws for logits
    k_layernorm<<<1, 256, 0, stream>>>(ws + WS_H, lnf_g, lnf_b, ws + WS_X,
                                       out + OUT_LAST);
    k_logits<<<129, 128, 0, stream>>>(ws + WS_X, lm_w, lscale, ws + WS_LOGITS);
    k_finalize<<<1, 256, 0, stream>>>(ws + WS_LOGITS, out);
}